// RGCNLayer_64854006169652
// MI455X (gfx1250) — compile-verified
//
#include <hip/hip_runtime.h>
#include <hip/hip_bf16.h>

#define N_NODES 50000
#define N_EDGES 100000
#define N_RELS  200
#define D       64
#define WAVES   4

typedef __attribute__((ext_vector_type(2))) float v2f;
typedef __attribute__((ext_vector_type(8))) float v8f;

// ---------------- init / binning ----------------

__global__ void k_zero(float* out, int* cnt, int* hist, int* cursor) {
    int i = blockIdx.x * blockDim.x + threadIdx.x;
    int stride = gridDim.x * blockDim.x;
    for (int j = i; j < N_NODES * D; j += stride) out[j] = 0.0f;
    for (int j = i; j < N_NODES; j += stride) cnt[j] = 0;
    for (int j = i; j < N_RELS; j += stride) { hist[j] = 0; cursor[j] = 0; }
}

__global__ void k_hist(const int* __restrict__ edges, int* hist, int* cnt) {
    int e = blockIdx.x * blockDim.x + threadIdx.x;
    if (e < N_EDGES) {
        atomicAdd(&hist[edges[e * 3 + 1]], 1);
        atomicAdd(&cnt[edges[e * 3 + 2]], 1);
    }
}

__global__ void k_scan(const int* __restrict__ hist, int* offs) {
    if (threadIdx.x == 0 && blockIdx.x == 0) {
        int acc = 0;
        for (int r = 0; r < N_RELS; ++r) { offs[r] = acc; acc += hist[r]; }
        offs[N_RELS] = acc;
    }
}

__global__ void k_scatter(const int* __restrict__ edges, const int* __restrict__ offs,
                          int* cursor, int* bucket) {
    int e = blockIdx.x * blockDim.x + threadIdx.x;
    if (e < N_EDGES) {
        int r = edges[e * 3 + 1];
        int pos = offs[r] + atomicAdd(&cursor[r], 1);
        bucket[pos] = e;
    }
}

// ---------------- edge messages: per-relation batched WMMA GEMM ----------------
// Block = 1 relation. B = W[rel] fragments held in registers for the whole block;
// A fragments loaded from LDS once per 16-edge tile; 64 WMMAs per tile run
// register-to-register. Padding rows scatter into a scrap sink (branch-free).

__global__ __launch_bounds__(128) void k_edge(const float* __restrict__ h,
                                              const float* __restrict__ W,
                                              const int* __restrict__ edges,
                                              const int* __restrict__ offs,
                                              const int* __restrict__ bucket,
                                              float* __restrict__ out,
                                              float* __restrict__ scrap) {
    __shared__ float sW[D * D];          // W[rel], row-major (staging only)
    __shared__ float sA[WAVES][16][68];  // stride 68 -> conflict-free 16x4 reads
    __shared__ int   sDst[WAVES][16];

    const int rel  = blockIdx.x;
    const int tid  = threadIdx.x;
    const int wave = tid >> 5;
    const int lane = tid & 31;
    const int base = offs[rel];
    const int count = offs[rel + 1] - base;

    const float* wsrc = W + (size_t)rel * D * D;
    for (int i = tid * 4; i < D * D; i += 128 * 4)
        *(float4*)&sW[i] = *(const float4*)&wsrc[i];
    __syncthreads();

    const int m     = lane & 15;
    const int khalf = (lane >> 4) * 2;   // lanes 0-15: K=0,1 ; lanes 16-31: K=2,3

    // preload ALL B fragments into registers: block-invariant (128 VGPRs)
    v2f bfrag[4][16];
    #pragma unroll
    for (int nt = 0; nt < 4; ++nt) {
        const int n = nt * 16 + m;
        #pragma unroll
        for (int k4 = 0; k4 < 16; ++k4) {
            const int k0 = k4 * 4 + khalf;
            bfrag[nt][k4].x = sW[k0 * D + n];
            bfrag[nt][k4].y = sW[(k0 + 1) * D + n];
        }
    }

    const int tiles = (count + 15) >> 4;
    float* const scrapRowBase = scrap + (size_t)(wave * 16) * D + m;

    for (int t0 = 0; t0 < tiles; t0 += WAVES) {   // uniform trip count per block
        const int t = t0 + wave;
        const bool activeWave = (t < tiles);
        if (activeWave) {
            // gather 16 edge rows into LDS
            int rowIdx  = t * 16 + m;
            int clamped = min(rowIdx, count - 1);
            int e    = bucket[base + clamped];
            int srcN = edges[e * 3 + 0];
            int dstN = edges[e * 3 + 2];
            if (lane < 16) sDst[wave][lane] = (rowIdx < count) ? dstN : -1;
            int half = lane >> 4;
            const float* hp = h + (size_t)srcN * D + half * 32;
            float* ap = &sA[wave][m][half * 32];
            #pragma unroll
            for (int j = 0; j < 32; j += 4)
                *(float4*)&ap[j] = *(const float4*)&hp[j];
        }
        __syncthreads();
        if (activeWave) {
            // A fragments: nt-invariant, load once per tile (32 VGPRs)
            v2f afrag[16];
            #pragma unroll
            for (int k4 = 0; k4 < 16; ++k4)
                afrag[k4] = *(const v2f*)&sA[wave][m][k4 * 4 + khalf];

            // destination pointers once per tile; invalid rows -> scrap sink
            float* addr[8];
            #pragma unroll
            for (int i = 0; i < 8; ++i) {
                int row = (lane < 16) ? i : (8 + i);       // C/D VGPR layout
                int dn = sDst[wave][row];
                addr[i] = (dn >= 0) ? (out + (size_t)dn * D + m)
                                    : (scrapRowBase + row * D);
            }

            #pragma unroll
            for (int nt = 0; nt < 4; ++nt) {
                v8f acc = {};
                #pragma unroll
                for (int k4 = 0; k4 < 16; ++k4)
                    acc = __builtin_amdgcn_wmma_f32_16x16x4_f32(
                        false, afrag[k4], false, bfrag[nt][k4],
                        (short)0, acc, false, false);
                #pragma unroll
                for (int i = 0; i < 8; ++i)
                    atomicAdd(addr[i] + nt * 16, acc[i]);   // nt*16 -> imm offset
            }
        }
        __syncthreads();
    }
}

// ---------------- self-loop GEMM + mean-combine ----------------

__global__ __launch_bounds__(128) void k_final(const float* __restrict__ h,
                                               const float* __restrict__ Wself,
                                               const int* __restrict__ cnt,
                                               float* __restrict__ out) {
    __shared__ float sW[D * D];
    const int tid = threadIdx.x, wave = tid >> 5, lane = tid & 31;
    for (int i = tid * 4; i < D * D; i += 128 * 4)
        *(float4*)&sW[i] = *(const float4*)&Wself[i];
    __syncthreads();

    const int rowTile = blockIdx.x * WAVES + wave;
    const int row0 = rowTile * 16;
    if (row0 >= N_NODES) return;

    const int m     = lane & 15;
    const int khalf = (lane >> 4) * 2;
    const int rowA  = min(row0 + m, N_NODES - 1);
    const float* arow = h + (size_t)rowA * D;
    const bool fullTile = (row0 + 16 <= N_NODES);

    // A fragments: direct global loads, one row per lane, nt-invariant
    v2f afrag[16];
    #pragma unroll
    for (int k4 = 0; k4 < 16; ++k4)
        afrag[k4] = *(const v2f*)&arow[k4 * 4 + khalf];

    // hoist per-row reciprocal degree and row pointers (one divide per row)
    float  invc[8];
    float* addr[8];
    #pragma unroll
    for (int i = 0; i < 8; ++i) {
        int rr = row0 + ((lane < 16) ? i : (8 + i));
        int rc = min(rr, N_NODES - 1);
        invc[i] = 1.0f / fmaxf((float)cnt[rc], 1.0f);
        addr[i] = out + (size_t)rc * D + m;
    }

    #pragma unroll
    for (int nt = 0; nt < 4; ++nt) {
        const int n = nt * 16 + m;
        // B fragments for this nt (register-resident across K chain)
        v2f bfrag[16];
        #pragma unroll
        for (int k4 = 0; k4 < 16; ++k4) {
            const int k0 = k4 * 4 + khalf;
            bfrag[k4].x = sW[k0 * D + n];
            bfrag[k4].y = sW[(k0 + 1) * D + n];
        }
        v8f acc = {};
        #pragma unroll
        for (int k4 = 0; k4 < 16; ++k4)
            acc = __builtin_amdgcn_wmma_f32_16x16x4_f32(
                false, afrag[k4], false, bfrag[k4], (short)0, acc, false, false);

        if (fullTile) {
            #pragma unroll
            for (int i = 0; i < 8; ++i) {
                float* p = addr[i] + nt * 16;
                *p = acc[i] + *p * invc[i];
            }
        } else {
            #pragma unroll
            for (int i = 0; i < 8; ++i) {
                int rr = row0 + ((lane < 16) ? i : (8 + i));
                if (rr < N_NODES) {
                    float* p = addr[i] + nt * 16;
                    *p = acc[i] + *p * invc[i];
                }
            }
        }
    }
}

// ---------------- launch ----------------

extern "C" void kernel_launch(void* const* d_in, const int* in_sizes, int n_in,
                              void* d_out, int out_size, void* d_ws, size_t ws_size,
                              hipStream_t stream) {
    const float* input_h = (const float*)d_in[0];
    const float* weight  = (const float*)d_in[1];
    const float* selfW   = (const float*)d_in[2];
    const int*   edges   = (const int*)d_in[3];
    float* out = (float*)d_out;

    int* cnt    = (int*)d_ws;            // N_NODES
    int* hist   = cnt + N_NODES;         // N_RELS
    int* offs   = hist + N_RELS;         // N_RELS + 1
    int* cursor = offs + N_RELS + 1;     // N_RELS
    int* bucket = cursor + N_RELS;       // N_EDGES
    float* scrap = (float*)(bucket + N_EDGES);  // WAVES*16*D floats, write-only sink

    k_zero<<<4096, 256, 0, stream>>>(out, cnt, hist, cursor);
    k_hist<<<(N_EDGES + 255) / 256, 256, 0, stream>>>(edges, hist, cnt);
    k_scan<<<1, 32, 0, stream>>>(hist, offs);
    k_scatter<<<(N_EDGES + 255) / 256, 256, 0, stream>>>(edges, offs, cursor, bucket);
    k_edge<<<N_RELS, 128, 0, stream>>>(input_h, weight, edges, offs, bucket, out, scrap);
    k_final<<<(N_NODES + 16 * WAVES - 1) / (16 * WAVES), 128, 0, stream>>>(
        input_h, selfW, cnt, out);
}